// LinearAttention_9646496546974
// MI455X (gfx1250) — compile-verified
//
#include <hip/hip_runtime.h>
#include <hip/hip_bf16.h>

typedef __attribute__((ext_vector_type(2))) float v2f;
typedef __attribute__((ext_vector_type(8))) float v8f;

#define BATCH   4
#define TOK     4096
#define HDIM    1024     // h * 64
#define NHEADS  16
#define ED      64       // head dim (e == d)
#define BH      (BATCH * NHEADS)     // 64
#define SPLITS  8                    // split-K blocks per head
#define CHUNKS  4                    // chunks per split
#define CHUNK   128                  // rows per chunk  (SPLITS*CHUNKS*CHUNK == TOK)
#define PKS     (SPLITS * 16)        // ksum partial slots per head (8 waves * 2 halves per split)

__device__ __forceinline__ float elu_fm(float x) {
    // elu(x) + 1  ==  x > 0 ? x + 1 : exp(x)
    return x > 0.0f ? x + 1.0f : expf(x);
}

// ---------------------------------------------------------------------------
// Pass 1: per (bh, split)  accumulate  context_partial[64][64] = K_fm^T * V
//         over 512 s-rows, plus per-lane k_sum partials.
// ---------------------------------------------------------------------------
__global__ __launch_bounds__(256) void la_ctx_partial(
    const float* __restrict__ key, const float* __restrict__ value,
    float* __restrict__ pctx, float* __restrict__ pksum)
{
    __shared__ float sK[CHUNK * ED];   // 32 KB, feature-mapped K chunk
    __shared__ float sV[CHUNK * ED];   // 32 KB

    const int bh    = blockIdx.x / SPLITS;
    const int split = blockIdx.x % SPLITS;
    const int b = bh >> 4, h = bh & 15;
    const int tid = threadIdx.x;
    const int wave = tid >> 5, lane = tid & 31;
    const int half = lane >> 4, lm = lane & 15;

    const size_t base = (size_t)b * TOK * HDIM + (size_t)h * ED;

    // tile pair for this wave: tiles (wave) and (wave+8); they share d0
    const int e0a = (wave >> 2) << 4;
    const int e0b = e0a + 32;
    const int d0  = (wave & 3) << 4;

    v8f acc0 = {};
    v8f acc1 = {};
    float4 kacc = make_float4(0.f, 0.f, 0.f, 0.f);

    for (int c = 0; c < CHUNKS; ++c) {
        __syncthreads();   // previous chunk's WMMA reads complete
        const int s0 = (split * CHUNKS + c) * CHUNK;
        // cooperative load: 128 rows * 16 float4 = 2048 float4s / 256 threads
        for (int i = tid; i < CHUNK * 16; i += 256) {
            const int row = i >> 4, c4 = i & 15;
            const size_t g = base + (size_t)(s0 + row) * HDIM + (size_t)c4 * 4;
            float4 kv = *(const float4*)(key + g);
            float4 vv = *(const float4*)(value + g);
            float4 kf;
            kf.x = elu_fm(kv.x); kf.y = elu_fm(kv.y);
            kf.z = elu_fm(kv.z); kf.w = elu_fm(kv.w);
            kacc.x += kf.x; kacc.y += kf.y; kacc.z += kf.z; kacc.w += kf.w;
            *(float4*)(sK + row * ED + c4 * 4) = kf;
            *(float4*)(sV + row * ED + c4 * 4) = vv;
        }
        __syncthreads();

        // 32 K-steps of 4 over this chunk; B (from V) shared by both tiles
        for (int s = 0; s < CHUNK; s += 4) {
            v2f a0, a1, bb;
            const int r0 = (s + 2 * half) * ED;
            const int r1 = (s + 1 + 2 * half) * ED;
            bb.x = sV[r0 + d0 + lm];
            bb.y = sV[r1 + d0 + lm];
            a0.x = sK[r0 + e0a + lm];
            a0.y = sK[r1 + e0a + lm];
            a1.x = sK[r0 + e0b + lm];
            a1.y = sK[r1 + e0b + lm];
            acc0 = __builtin_amdgcn_wmma_f32_16x16x4_f32(
                false, a0, false, bb, (short)0, acc0, false, false);
            acc1 = __builtin_amdgcn_wmma_f32_16x16x4_f32(
                false, a1, false, bb, (short)0, acc1, false, false);
        }
    }

    // ksum partial: lane covers columns 4*(lane&15)..+3; each (wave,half) gets
    // its own slot so no cross-lane reduction is needed (fixed-order later).
    {
        float* dst = pksum + ((size_t)bh * PKS + split * 16 + wave * 2 + half) * ED
                           + lm * 4;
        *(float4*)dst = kacc;
    }

    // write context partial tiles (C layout: VGPR r -> row e0+r+8*half)
    float* pc = pctx + ((size_t)bh * SPLITS + split) * (ED * ED);
    #pragma unroll
    for (int r = 0; r < 8; ++r) {
        pc[(e0a + r + 8 * half) * ED + d0 + lm] = acc0[r];
        pc[(e0b + r + 8 * half) * ED + d0 + lm] = acc1[r];
    }
}

// ---------------------------------------------------------------------------
// Pass 2: fixed-order reduction of partials -> context[bh][64][64], ksum[bh][64]
// ---------------------------------------------------------------------------
__global__ __launch_bounds__(256) void la_reduce(
    const float* __restrict__ pctx, const float* __restrict__ pksum,
    float* __restrict__ ctx, float* __restrict__ ksum)
{
    const int bh = blockIdx.x, tid = threadIdx.x;
    for (int j = tid; j < ED * ED; j += 256) {
        float s = 0.f;
        #pragma unroll
        for (int sp = 0; sp < SPLITS; ++sp)
            s += pctx[((size_t)bh * SPLITS + sp) * (ED * ED) + j];
        ctx[(size_t)bh * (ED * ED) + j] = s;
    }
    if (tid < ED) {
        float s = 0.f;
        for (int sl = 0; sl < PKS; ++sl)
            s += pksum[((size_t)bh * PKS + sl) * ED + tid];
        ksum[bh * ED + tid] = s;
    }
}

// ---------------------------------------------------------------------------
// Pass 3: out[n][d] = D_inv[n] * sum_e Q_fm[n][e] * context[e][d]
//         one block = one (bh) head x 64 token rows
// ---------------------------------------------------------------------------
__global__ __launch_bounds__(256) void la_out(
    const float* __restrict__ query, const float* __restrict__ ctx,
    const float* __restrict__ ksum, float* __restrict__ out)
{
    __shared__ float sQ[64 * ED];     // 16 KB, feature-mapped Q rows
    __shared__ float sC[ED * ED];     // 16 KB
    __shared__ float sKs[ED];
    __shared__ float sDinv[64];

    const int bh = blockIdx.y;
    const int b = bh >> 4, h = bh & 15;
    const int t0 = blockIdx.x * 64;
    const int tid = threadIdx.x;
    const int wave = tid >> 5, lane = tid & 31;
    const int half = lane >> 4, lm = lane & 15;

    const size_t qbase = (size_t)b * TOK * HDIM + (size_t)h * ED;

    for (int i = tid; i < 64 * 16; i += 256) {
        const int row = i >> 4, c4 = i & 15;
        *(float4*)(sC + row * ED + c4 * 4) =
            *(const float4*)(ctx + (size_t)bh * (ED * ED) + row * ED + c4 * 4);
        float4 qv = *(const float4*)(query + qbase + (size_t)(t0 + row) * HDIM
                                     + (size_t)c4 * 4);
        float4 qf;
        qf.x = elu_fm(qv.x); qf.y = elu_fm(qv.y);
        qf.z = elu_fm(qv.z); qf.w = elu_fm(qv.w);
        *(float4*)(sQ + row * ED + c4 * 4) = qf;
    }
    if (tid < ED) sKs[tid] = ksum[bh * ED + tid];
    __syncthreads();

    if (tid < 64) {
        float d = 0.f;
        #pragma unroll 8
        for (int e = 0; e < ED; ++e) d += sQ[tid * ED + e] * sKs[e];
        sDinv[tid] = 1.0f / (d + 1e-6f);
    }
    __syncthreads();

    const int n0a = (wave >> 2) << 4;
    const int n0b = n0a + 32;
    const int d0  = (wave & 3) << 4;

    v8f acc0 = {};
    v8f acc1 = {};
    for (int e = 0; e < ED; e += 4) {
        v2f a0, a1, bb;
        bb.x = sC[(e + 2 * half) * ED + d0 + lm];
        bb.y = sC[(e + 1 + 2 * half) * ED + d0 + lm];
        a0.x = sQ[(n0a + lm) * ED + e + 2 * half];
        a0.y = sQ[(n0a + lm) * ED + e + 1 + 2 * half];
        a1.x = sQ[(n0b + lm) * ED + e + 2 * half];
        a1.y = sQ[(n0b + lm) * ED + e + 1 + 2 * half];
        acc0 = __builtin_amdgcn_wmma_f32_16x16x4_f32(
            false, a0, false, bb, (short)0, acc0, false, false);
        acc1 = __builtin_amdgcn_wmma_f32_16x16x4_f32(
            false, a1, false, bb, (short)0, acc1, false, false);
    }

    #pragma unroll
    for (int r = 0; r < 8; ++r) {
        const int rowa = n0a + r + 8 * half;
        const int rowb = n0b + r + 8 * half;
        out[((size_t)b * TOK + t0 + rowa) * HDIM + h * ED + d0 + lm]
            = acc0[r] * sDinv[rowa];
        out[((size_t)b * TOK + t0 + rowb) * HDIM + h * ED + d0 + lm]
            = acc1[r] * sDinv[rowb];
    }
}

// ---------------------------------------------------------------------------
extern "C" void kernel_launch(void* const* d_in, const int* in_sizes, int n_in,
                              void* d_out, int out_size, void* d_ws, size_t ws_size,
                              hipStream_t stream) {
    (void)in_sizes; (void)n_in; (void)out_size; (void)ws_size;
    const float* q = (const float*)d_in[0];
    const float* k = (const float*)d_in[1];
    const float* v = (const float*)d_in[2];
    float* out = (float*)d_out;

    float* wsf   = (float*)d_ws;
    float* pctx  = wsf;                                        // 64*8*4096 f32
    float* pksum = pctx  + (size_t)BH * SPLITS * (ED * ED);    // 64*128*64 f32
    float* ctx   = pksum + (size_t)BH * PKS * ED;              // 64*4096 f32
    float* ksum  = ctx   + (size_t)BH * (ED * ED);             // 64*64 f32

    la_ctx_partial<<<BH * SPLITS, 256, 0, stream>>>(k, v, pctx, pksum);
    la_reduce<<<BH, 256, 0, stream>>>(pctx, pksum, ctx, ksum);
    la_out<<<dim3(TOK / 64, BH), 256, 0, stream>>>(q, ctx, ksum, out);
}